// GCNCustomConv_51788715655556
// MI455X (gfx1250) — compile-verified
//
#include <hip/hip_runtime.h>

typedef __attribute__((ext_vector_type(2))) float v2f;
typedef __attribute__((ext_vector_type(8))) float v8f;

#define D 128  // D_IN == D_OUT == 128 per reference

// ---- 1. degree init: every node gets +1 from its self-loop -----------------
__global__ void k_deg_init(float* __restrict__ deg, int n) {
    int i = blockIdx.x * blockDim.x + threadIdx.x;
    if (i < n) deg[i] = 1.0f;
}

// ---- 2. degree accumulate over col = edge_index[1] -------------------------
__global__ void k_deg_accum(const long long* __restrict__ ei, float* __restrict__ deg, int nE) {
    int e = blockIdx.x * blockDim.x + threadIdx.x;
    if (e < nE) {
        int c = (int)ei[(size_t)nE + e];
        atomicAdd(&deg[c], 1.0f);
    }
}

// ---- 3. dis = deg > 0 ? rsqrt(deg) : 0 -------------------------------------
__global__ void k_rsqrt(const float* __restrict__ deg, float* __restrict__ dis, int n) {
    int i = blockIdx.x * blockDim.x + threadIdx.x;
    if (i < n) {
        float d = deg[i];
        dis[i] = (d > 0.0f) ? rsqrtf(d) : 0.0f;
    }
}

// ---- 4. h = x @ W via V_WMMA_F32_16X16X4_F32 -------------------------------
// Block = 256 threads = 8 waves. Wave w computes the 16x16 tile at
// (m0 = 16*blockIdx.x, n0 = 16*w). K loop fully unrolled: 32 WMMAs per wave.
// A 16x4 layout: lanes 0-15 hold K=0,1 (2 VGPRs), lanes 16-31 hold K=2,3.
// B 4x16 mirrored; C/D: VGPR i -> M = i (+8 for upper lane half), N = lane%16.
__global__ void __launch_bounds__(256)
k_gemm_wmma(const float* __restrict__ x, const float* __restrict__ W,
            float* __restrict__ h, int nRows) {
    const int wave  = threadIdx.x >> 5;
    const int lane  = threadIdx.x & 31;
    const int lrow  = lane & 15;
    const int khalf = lane >> 4;           // 0: K=0,1  1: K=2,3
    const int m0 = blockIdx.x * 16;
    const int n0 = wave * 16;

    int arow = m0 + lrow;
    if (arow >= nRows) arow = nRows - 1;   // clamp loads; EXEC stays all-1s for WMMA

    const float* aBase = x + (size_t)arow * D + 2 * khalf;
    const float* bBase = W + (size_t)(2 * khalf) * D + n0 + lrow;

    v8f acc = {};
#pragma unroll
    for (int k = 0; k < D; k += 4) {
        v2f a, b;
        a.x = aBase[k + 0];                // A[arow, k + 2*khalf + 0]
        a.y = aBase[k + 1];                // A[arow, k + 2*khalf + 1]
        b.x = bBase[(size_t)(k + 0) * D];  // W[k + 2*khalf + 0, n0 + lrow]
        b.y = bBase[(size_t)(k + 1) * D];  // W[k + 2*khalf + 1, n0 + lrow]
        acc = __builtin_amdgcn_wmma_f32_16x16x4_f32(
            /*neg_a=*/false, a, /*neg_b=*/false, b,
            /*c_mod=*/(short)0, acc, /*reuse_a=*/false, /*reuse_b=*/false);
    }

#pragma unroll
    for (int i = 0; i < 8; ++i) {
        int r = m0 + khalf * 8 + i;
        if (r < nRows) h[(size_t)r * D + n0 + lrow] = acc[i];
    }
}

// ---- 5. out = bias + dis^2 * h  (self-loop term; also initializes d_out) ---
__global__ void k_init_out(const float* __restrict__ h, const float* __restrict__ dis,
                           const float* __restrict__ bias, float* __restrict__ out, int total) {
    int i = blockIdx.x * blockDim.x + threadIdx.x;
    if (i < total) {
        int node = i >> 7;                 // / 128
        int d    = i & (D - 1);
        float s  = dis[node];
        out[i] = bias[d] + s * s * h[i];
    }
}

// ---- 6. edge scatter: out[row] += dis[row]*dis[col] * h[col] ---------------
// One wave per edge; each lane handles 4 consecutive dims (float4 gather,
// 4x global_atomic_add_f32). h and out both fit in the 192MB L2, so the
// random traffic and atomics are serviced at L2 bandwidth.
__global__ void __launch_bounds__(256)
k_scatter(const long long* __restrict__ ei, const float* __restrict__ h,
          const float* __restrict__ dis, float* __restrict__ out, int nE) {
    const int wavesPerBlock = blockDim.x >> 5;
    const int e = blockIdx.x * wavesPerBlock + (threadIdx.x >> 5);
    if (e >= nE) return;
    const int lane = threadIdx.x & 31;

    int r = (int)ei[e];                    // row = edge_index[0][e]
    int c = (int)ei[(size_t)nE + e];       // col = edge_index[1][e]
    float norm = dis[r] * dis[c];

    const float4* hc = (const float4*)(h + (size_t)c * D);
    float4 v = hc[lane];

    float* o = out + (size_t)r * D + lane * 4;
    atomicAdd(o + 0, norm * v.x);
    atomicAdd(o + 1, norm * v.y);
    atomicAdd(o + 2, norm * v.z);
    atomicAdd(o + 3, norm * v.w);
}

extern "C" void kernel_launch(void* const* d_in, const int* in_sizes, int n_in,
                              void* d_out, int out_size, void* d_ws, size_t ws_size,
                              hipStream_t stream) {
    const float*     x    = (const float*)d_in[0];       // [N, 128]
    const float*     W    = (const float*)d_in[1];       // [128, 128]
    const float*     bias = (const float*)d_in[2];       // [128]
    const long long* ei   = (const long long*)d_in[3];   // [2, E] int64

    const int N = in_sizes[0] / D;
    const int E = in_sizes[3] / 2;

    float* out = (float*)d_out;

    // workspace layout: h [N*128] | deg [N] | dis [N]  (~25.8 MB total)
    float* h   = (float*)d_ws;
    float* deg = h + (size_t)N * D;
    float* dis = deg + N;

    k_deg_init <<<(N + 255) / 256, 256, 0, stream>>>(deg, N);
    k_deg_accum<<<(E + 255) / 256, 256, 0, stream>>>(ei, deg, E);
    k_rsqrt    <<<(N + 255) / 256, 256, 0, stream>>>(deg, dis, N);
    k_gemm_wmma<<<(N + 15) / 16, 256, 0, stream>>>(x, W, h, N);
    int total = N * D;
    k_init_out <<<(total + 255) / 256, 256, 0, stream>>>(h, dis, bias, out, total);
    k_scatter  <<<(E + 7) / 8, 256, 0, stream>>>(ei, h, dis, out, E);
}